// Aggregator_64750926954866
// MI455X (gfx1250) — compile-verified
//
#include <hip/hip_runtime.h>

// GNN aggregator for MI455X (gfx1250, wave32):
//   agg[dst] += x[src] * edge_attr   (scatter, atomic f32)
//   out = LeakyReLU(agg @ W.T + b)   (WMMA f32 16x16x4 GEMM)

#define IN_DIM   128
#define OUT_DIM  128
#define SLOPE    0.01f

typedef float v2f __attribute__((ext_vector_type(2)));
typedef float v8f __attribute__((ext_vector_type(8)));

// ---------------------------------------------------------------------------
// Kernel 1: zero the aggregation workspace (float4 stores, grid-stride)
// ---------------------------------------------------------------------------
__global__ __launch_bounds__(256) void agg_zero_kernel(float4* __restrict__ agg,
                                                       long long n4) {
    long long i = (long long)blockIdx.x * blockDim.x + threadIdx.x;
    long long stride = (long long)gridDim.x * blockDim.x;
    float4 z = make_float4(0.f, 0.f, 0.f, 0.f);
    for (; i < n4; i += stride) agg[i] = z;
}

// ---------------------------------------------------------------------------
// Kernel 2: one wave32 per edge. Lane L handles features [4L, 4L+4).
//   - float4 coalesced read of x[src] row (512B per wave)
//   - 4x global_atomic_add_f32 into agg[dst] row
// ---------------------------------------------------------------------------
__global__ __launch_bounds__(256) void edge_scatter_kernel(
        const float* __restrict__ x,
        const int*   __restrict__ src,
        const int*   __restrict__ dst,
        const float* __restrict__ edge_attr,
        float*       __restrict__ agg,
        int n_edges) {
    const int lane = threadIdx.x & 31;
    const int e    = blockIdx.x * 8 + (threadIdx.x >> 5);   // 8 waves per block
    if (e >= n_edges) return;

    const int   s = src[e];
    const int   d = dst[e];
    const float a = edge_attr[e];

    const float4 v = *(const float4*)(x + (size_t)s * IN_DIM + lane * 4);
    float* p = agg + (size_t)d * IN_DIM + lane * 4;
    atomicAdd(p + 0, v.x * a);
    atomicAdd(p + 1, v.y * a);
    atomicAdd(p + 2, v.z * a);
    atomicAdd(p + 3, v.w * a);
}

// ---------------------------------------------------------------------------
// Kernel 3: out = LeakyReLU(agg @ W.T + b) via V_WMMA_F32_16X16X4_F32.
// One wave per 16x16 output tile; block = 8 waves = 1 M-tile x 8 N-tiles.
// K = 128 -> chain of 32 WMMAs accumulating in the v8f C registers.
//
// Layouts (wave32, from CDNA5 ISA):
//   A 16x4 : lane m = lane&15, kbase = (lane>>4)*2; VGPR0/1 = A[m][kbase+0/1]
//   B 4x16 : lane n = lane&15, same K split;        B[k][n] = W[n][k]
//   C/D    : VGPR r -> M = (lane>>4)*8 + r, N = lane&15
// ---------------------------------------------------------------------------
__global__ __launch_bounds__(256) void gemm_bias_lrelu_kernel(
        const float* __restrict__ agg,
        const float* __restrict__ W,     // [OUT_DIM, IN_DIM] row-major
        const float* __restrict__ bias,  // [OUT_DIM]
        float*       __restrict__ out) {
    const int lane  = threadIdx.x & 31;
    const int wave  = threadIdx.x >> 5;     // 0..7 -> N tile
    const int mtile = blockIdx.x;           // 0..6249
    const int ntile = wave;                 // 0..7  (OUT_DIM/16)

    const int m     = lane & 15;
    const int n     = lane & 15;
    const int kbase = (lane >> 4) << 1;     // 0 or 2 within each K=4 step

    const float* arow = agg + (size_t)(mtile * 16 + m) * IN_DIM;
    const float* wrow = W   + (size_t)(ntile * 16 + n) * IN_DIM; // row n of W == col n of W.T

    v8f c = {};
#pragma unroll
    for (int k0 = 0; k0 < IN_DIM; k0 += 4) {
        v2f a = *(const v2f*)(arow + k0 + kbase);
        v2f b = *(const v2f*)(wrow + k0 + kbase);
        // 8 args: (neg_a, A, neg_b, B, c_mod, C, reuse_a, reuse_b)
        c = __builtin_amdgcn_wmma_f32_16x16x4_f32(
                false, a, false, b, (short)0, c, false, false);
    }

    const int col     = ntile * 16 + (lane & 15);
    const int rowbase = mtile * 16 + ((lane >> 4) << 3);
    const float bb    = bias[col];
#pragma unroll
    for (int r = 0; r < 8; ++r) {
        float v = c[r] + bb;
        out[(size_t)(rowbase + r) * OUT_DIM + col] = (v >= 0.f) ? v : SLOPE * v;
    }
}

// ---------------------------------------------------------------------------
// Host-side launch
// ---------------------------------------------------------------------------
extern "C" void kernel_launch(void* const* d_in, const int* in_sizes, int n_in,
                              void* d_out, int out_size, void* d_ws, size_t ws_size,
                              hipStream_t stream) {
    const float* x  = (const float*)d_in[0];        // [N, 128]
    const int*   ei = (const int*)  d_in[1];        // [2, E] flat: src then dst
    const float* ea = (const float*)d_in[2];        // [E]
    const float* W  = (const float*)d_in[3];        // [128, 128]
    const float* b  = (const float*)d_in[4];        // [128]
    float*       out = (float*)d_out;
    float*       agg = (float*)d_ws;                // [N, 128] accumulator

    const int E = in_sizes[1] / 2;
    const int N = out_size / OUT_DIM;               // 100000 (divisible by 16)

    // 1) zero accumulator (must happen every launch: d_ws is not re-poisoned)
    {
        long long n4 = (long long)N * IN_DIM / 4;
        int blocks = (int)((n4 + 255) / 256);
        if (blocks > 4096) blocks = 4096;           // grid-stride
        agg_zero_kernel<<<blocks, 256, 0, stream>>>((float4*)agg, n4);
    }

    // 2) scatter: one wave per edge, 8 waves per block
    {
        int blocks = (E + 7) / 8;
        edge_scatter_kernel<<<blocks, 256, 0, stream>>>(x, ei, ei + E, ea, agg, E);
    }

    // 3) GEMM + bias + LeakyReLU: one block per 16-row M tile
    {
        int blocks = N / 16;                        // 6250
        gemm_bias_lrelu_kernel<<<blocks, 256, 0, stream>>>(agg, W, b, out);
    }
}